// ProtoTriplet_61521111548487
// MI455X (gfx1250) — compile-verified
//
#include <hip/hip_runtime.h>

typedef float v2f __attribute__((ext_vector_type(2)));
typedef float v4f __attribute__((ext_vector_type(4)));
typedef float v8f __attribute__((ext_vector_type(8)));

#define B_ROWS   262144
#define D_DIM    128
#define C_PROTO  100
#define NPAD     112            // 7 N-tiles of 16
#define NT       7
#define LDSTR    132            // padded LDS row stride (dwords) -> conflict-light
#define MARGIN   0.5f
#define EPS_     1.0e-6f
#define D_EPS2   (128.0f * 1.0e-12f)
#define PAD_P2   16.0f          // padded protos: dist ~ sqrt(17) -> hinge == 0, maskless

#define BLOCKS   256
#define THREADS  256
#define WAVES_PER_BLOCK 8
#define TOTAL_WAVES (BLOCKS * WAVES_PER_BLOCK)   // 2048
#define TILES    (B_ROWS / 16)                   // 16384 -> 8 tiles per wave exactly

// ---------------------------------------------------------------------------
// Stage 1: normalize prototypes, pad to 112 rows, precompute p2 / sp
// ---------------------------------------------------------------------------
__global__ void proto_prep_kernel(const float* __restrict__ p,
                                  float* __restrict__ gp,
                                  float* __restrict__ gp2,
                                  float* __restrict__ gsp) {
    int r = blockIdx.x * blockDim.x + threadIdx.x;
    if (r >= NPAD) return;
    if (r < C_PROTO) {
        float ss = 0.f, s = 0.f;
        for (int c = 0; c < D_DIM; ++c) {
            float x = p[r * D_DIM + c];
            ss += x * x;
            s  += x;
        }
        float rinv = 1.0f / fmaxf(sqrtf(ss), 1e-12f);
        for (int c = 0; c < D_DIM; ++c)
            gp[r * D_DIM + c] = p[r * D_DIM + c] * rinv;
        gp2[r] = ss * rinv * rinv;   // == sum(p_norm^2)
        gsp[r] = s * rinv;           // == sum(p_norm)
    } else {
        for (int c = 0; c < D_DIM; ++c) gp[r * D_DIM + c] = 0.f;
        gp2[r] = PAD_P2;             // pushes padded distances above any d_pos+margin
        gsp[r] = 0.f;
    }
}

// ---------------------------------------------------------------------------
// Stage 2: fused normalize + GEMM (WMMA f32 16x16x4) + hinge reduction
// ---------------------------------------------------------------------------
__global__ void __launch_bounds__(THREADS)
proto_triplet_main_kernel(const float* __restrict__ f,
                          const int*  __restrict__ labels,
                          const float* __restrict__ gp,
                          const float* __restrict__ gp2,
                          const float* __restrict__ gsp,
                          float* __restrict__ partials) {
    __shared__ float s_p[NPAD * LDSTR];
    __shared__ float s_p2[NPAD];
    __shared__ float s_sp[NPAD];
    __shared__ float s_wsum[WAVES_PER_BLOCK];

    const int tid   = threadIdx.x;
    const int wave  = tid >> 5;
    const int lane  = tid & 31;
    const int lr    = lane & 15;       // row / proto index within tile
    const int half  = lane >> 4;       // K-half selector

    // --- cooperative prototype stage into LDS (padded rows) ---
    for (int i = tid; i < NPAD * D_DIM; i += THREADS) {
        int r = i >> 7, c = i & (D_DIM - 1);
        s_p[r * LDSTR + c] = gp[i];
    }
    if (tid < NPAD) {
        s_p2[tid] = gp2[tid];
        s_sp[tid] = gsp[tid];
    }
    __syncthreads();

    // per-lane folded constants for the 7 N-tiles (invariant across tiles):
    // base[t] = p2[j] - 2*EPS*sp[j] + D*EPS^2
    float base[NT];
#pragma unroll
    for (int t = 0; t < NT; ++t)
        base[t] = s_p2[t * 16 + lr] - 2.0f * EPS_ * s_sp[t * 16 + lr] + D_EPS2;

    const int gwave = blockIdx.x * WAVES_PER_BLOCK + wave;
    float lsum = 0.f;

    for (int tile = gwave; tile < TILES; tile += TOTAL_WAVES) {
        const int   rowbase = tile * 16;
        const float* fp = f + (size_t)(rowbase + lr) * D_DIM + half * 4;
        const int   labl = labels[rowbase + lr];

        v8f acc[NT];
#pragma unroll
        for (int t = 0; t < NT; ++t)
            acc[t] = (v8f){0.f, 0.f, 0.f, 0.f, 0.f, 0.f, 0.f, 0.f};

        float ssq = 0.f, ssm = 0.f;

        // 16 macro k-steps; keep ROLLED (unroll 2) to avoid the >256-VGPR
        // hoist + scratch spills seen with full unroll.
#pragma unroll 2
        for (int km = 0; km < 16; ++km) {
            v4f a4 = *(const v4f*)(fp + km * 8);
            ssq += a4.x * a4.x + a4.y * a4.y + a4.z * a4.z + a4.w * a4.w;
            ssm += a4.x + a4.y + a4.z + a4.w;
            v2f a0 = {a4.x, a4.y};
            v2f a1 = {a4.z, a4.w};

            v4f b4[NT];
#pragma unroll
            for (int t = 0; t < NT; ++t)
                b4[t] = *(const v4f*)&s_p[(t * 16 + lr) * LDSTR + km * 8 + half * 4];
#pragma unroll
            for (int t = 0; t < NT; ++t) {
                v2f b0 = {b4[t].x, b4[t].y};
                v2f b1 = {b4[t].z, b4[t].w};
                acc[t] = __builtin_amdgcn_wmma_f32_16x16x4_f32(
                    false, a0, false, b0, (short)0, acc[t], false, false);
                acc[t] = __builtin_amdgcn_wmma_f32_16x16x4_f32(
                    false, a1, false, b1, (short)0, acc[t], false, false);
            }
        }

        // complete row reductions: lane l and l^16 hold complementary K halves
        ssq += __shfl_xor(ssq, 16);
        ssm += __shfl_xor(ssm, 16);
        float rinv = 1.0f / fmaxf(sqrtf(ssq), 1e-12f);
        float cl   = ssq * rinv * rinv + 2.0f * EPS_ * (ssm * rinv); // f2 + 2*EPS*sf
        float m2l  = -2.0f * rinv;

        const int srcBase = half << 3;    // lanes>=16 own rows 8..15
#pragma unroll
        for (int v = 0; v < 8; ++v) {
            const int srcM = v + srcBase;            // row M for this VGPR/half
            float m2v  = __shfl(m2l, srcM);
            float cv   = __shfl(cl, srcM);
            int   labv = __shfl(labl, srcM);

            float dposv = 0.f;
            float dv[NT];
#pragma unroll
            for (int t = 0; t < NT; ++t) {
                float d2   = fmaf(acc[t][v], m2v, base[t] + cv);
                float dist = sqrtf(fmaxf(d2, 1e-12f));
                dv[t] = dist;
                dposv += (t * 16 + lr == labv) ? dist : 0.f;
            }
            // exactly one lane in this half matched the label: sum-broadcast
            dposv += __shfl_xor(dposv, 1);
            dposv += __shfl_xor(dposv, 2);
            dposv += __shfl_xor(dposv, 4);
            dposv += __shfl_xor(dposv, 8);

            // maskless hinge: padded cols give 0 (dist >= 4 > hv), diagonal
            // contributes exactly MARGIN, corrected analytically at the end.
            float hv = dposv + MARGIN;
#pragma unroll
            for (int t = 0; t < NT; ++t)
                lsum += fmaxf(0.f, hv - dv[t]);
        }
    }

    // deterministic wave -> block reduction
    lsum += __shfl_xor(lsum, 1);
    lsum += __shfl_xor(lsum, 2);
    lsum += __shfl_xor(lsum, 4);
    lsum += __shfl_xor(lsum, 8);
    lsum += __shfl_xor(lsum, 16);
    if (lane == 0) s_wsum[wave] = lsum;
    __syncthreads();
    if (tid == 0) {
        float s = 0.f;
        for (int w = 0; w < WAVES_PER_BLOCK; ++w) s += s_wsum[w];
        partials[blockIdx.x] = s;
    }
}

// ---------------------------------------------------------------------------
// Stage 3: deterministic final reduction (fp64; removes diagonal MARGIN terms)
// ---------------------------------------------------------------------------
__global__ void finalize_kernel(const float* __restrict__ partials,
                                float* __restrict__ out, int n) {
    if (threadIdx.x == 0 && blockIdx.x == 0) {
        double s = 0.0;
        for (int i = 0; i < n; ++i) s += (double)partials[i];
        s -= (double)MARGIN * (double)B_ROWS;        // remove j==label terms
        out[0] = (float)(s / ((double)(C_PROTO - 1) * (double)B_ROWS));
    }
}

// ---------------------------------------------------------------------------
extern "C" void kernel_launch(void* const* d_in, const int* in_sizes, int n_in,
                              void* d_out, int out_size, void* d_ws, size_t ws_size,
                              hipStream_t stream) {
    (void)in_sizes; (void)n_in; (void)out_size; (void)ws_size;
    const float* features   = (const float*)d_in[0];
    const float* prototypes = (const float*)d_in[1];
    const int*   labels     = (const int*)d_in[2];

    float* gp       = (float*)d_ws;            // [112 * 128] normalized prototypes
    float* gp2      = gp + NPAD * D_DIM;       // [112]
    float* gsp      = gp2 + NPAD;              // [112]
    float* partials = gsp + NPAD;              // [BLOCKS]

    proto_prep_kernel<<<1, 128, 0, stream>>>(prototypes, gp, gp2, gsp);
    proto_triplet_main_kernel<<<BLOCKS, THREADS, 0, stream>>>(
        features, labels, gp, gp2, gsp, partials);
    finalize_kernel<<<1, 32, 0, stream>>>(partials, (float*)d_out, BLOCKS);
}